// CharacterLevelWordSparseEncoding_31868657336780
// MI455X (gfx1250) — compile-verified
//
#include <hip/hip_runtime.h>

// Bag-of-characters histogram: out[b,w,v] = count of v in token_ids[b,w,:], out[:,:,0]=0.
// B=64, W=512, L=24, V=1024. Pure store-bandwidth-bound (134 MB out).
// CDNA5 path: build each 4KB row in LDS (ds_store_b128 zero-fill + ds_add_f32
// scatter), then DMA it out with global_store_async_from_lds_b128 (ASYNCcnt).

#define VOCAB 1024
#define WLEN 24
#define WAVES_PER_BLOCK 8

__global__ __launch_bounds__(256) void bag_of_chars_kernel(
    const int* __restrict__ tokens, float* __restrict__ out, int nrows) {
  // 8 waves per block, one 4KB histogram row per wave. 16B-aligned for b128 DS ops.
  __shared__ __align__(16) float hist[WAVES_PER_BLOCK][VOCAB];

  const int tid  = (int)threadIdx.x;
  const int wave = tid >> 5;   // wave32
  const int lane = tid & 31;
  const int row  = (int)blockIdx.x * WAVES_PER_BLOCK + wave;
  if (row >= nrows) return;    // uniform per wave

  float* h = &hist[wave][0];

  // ---- 1) Zero the 4KB row: 8 x ds_store_b128 per lane (32 lanes x 128B) ----
  float4* hv4 = reinterpret_cast<float4*>(h);
  const float4 z = make_float4(0.f, 0.f, 0.f, 0.f);
#pragma unroll
  for (int i = 0; i < 8; ++i) hv4[lane + 32 * i] = z;

  // ---- 2) Scatter-add the 24 tokens. ds_add_f32 handles intra-row duplicates;
  //         LDS ops from one wave are in-order, so no barrier vs the zero-fill.
  //         Skip padding idx 0 entirely (reference zeroes that column). ----
  if (lane < WLEN) {
    int t = tokens[(size_t)row * WLEN + lane];
    if (t != 0) atomicAdd(&h[t], 1.0f);
  }

  // ---- 3) Drain the DS pipe so the async engine reads completed LDS data ----
  asm volatile("s_wait_dscnt 0" ::: "memory");

  // ---- 4) Async LDS -> global: 8 x b128 per lane = 4KB contiguous per wave.
  //         The instruction IOFFSET applies to BOTH the LDS source and the
  //         global destination (CDNA5 ISA), so one base pair
  //         + offset:i*512 covers the whole row. No explicit asynccnt wait is
  //         needed: each wave does exactly one row and S_ENDPGM waits idle. ----
  // Low 32 bits of the flat shared-aperture address == LDS byte offset.
  unsigned lds_byte =
      (unsigned)(unsigned long long)h + (unsigned)(lane * 16);
  unsigned long long gaddr =
      (unsigned long long)(out + (size_t)row * VOCAB + lane * 4);
#pragma unroll
  for (int i = 0; i < 8; ++i) {
    asm volatile("global_store_async_from_lds_b128 %0, %1, off offset:%2"
                 :
                 : "v"(gaddr), "v"(lds_byte), "i"(i * 512)
                 : "memory");
  }
}

extern "C" void kernel_launch(void* const* d_in, const int* in_sizes, int n_in,
                              void* d_out, int out_size, void* d_ws, size_t ws_size,
                              hipStream_t stream) {
  const int* tokens = (const int*)d_in[0];
  float* out = (float*)d_out;

  const int nrows = in_sizes[0] / WLEN;  // 64*512 = 32768 rows
  const int blocks = (nrows + WAVES_PER_BLOCK - 1) / WAVES_PER_BLOCK;  // 4096

  hipLaunchKernelGGL(bag_of_chars_kernel, dim3(blocks), dim3(256), 0, stream,
                     tokens, out, nrows);
}